// EdgeConv_90142773609278
// MI455X (gfx1250) — compile-verified
//
#include <hip/hip_runtime.h>
#include <hip/hip_bf16.h>

#define B_ 4
#define C_ 64
#define N_ 4096
#define K_ 20
#define OUT_ 64
#define IN_ 128
#define EPS_ 1e-5f
#define SLOPE_ 0.2f

typedef __attribute__((ext_vector_type(2))) float v2f;
typedef __attribute__((ext_vector_type(8))) float v8f;

// ---------------------------------------------------------------------------
// Kernel A: xx[b][n] = sum_c x[b][c][n]^2
// ---------------------------------------------------------------------------
__launch_bounds__(256)
__global__ void k_xx(const float* __restrict__ x, float* __restrict__ xx) {
  int gid = blockIdx.x * blockDim.x + threadIdx.x;   // 0 .. B*N-1
  int b = gid / N_, n = gid - b * N_;
  const float* xb = x + (size_t)b * C_ * N_;
  float s = 0.f;
  #pragma unroll 8
  for (int c = 0; c < C_; ++c) { float v = xb[c * N_ + n]; s += v * v; }
  xx[gid] = s;
}

// ---------------------------------------------------------------------------
// Kernel B: fused distance-GEMM (WMMA f32 16x16x4) + per-row top-20 selection.
// One wave (32 threads) owns 32 rows; sweeps all 256 column tiles of 16.
// neg_dist[m][n] = 2*inner - xx[m] - xx[n]; self-match (dist 0) kept, matching
// the reference top_k semantics. k-order is irrelevant downstream (mean/var/max).
// ---------------------------------------------------------------------------
__launch_bounds__(32)
__global__ void k_knn(const float* __restrict__ x, const float* __restrict__ xx,
                      int* __restrict__ knn) {
  __shared__ float s_tile[32][17];   // padded to dodge bank conflicts
  __shared__ float s_val[32][K_];
  __shared__ int   s_id[32][K_];

  const int lane = threadIdx.x;
  const int half = lane >> 4;        // 0: K=0,1  1: K=2,3 (A/B frag layout)
  const int l16  = lane & 15;
  const int blk  = blockIdx.x;       // 0 .. B*(N/32)-1
  const int b    = blk / (N_ / 32);
  const int r0   = (blk - b * (N_ / 32)) * 32;
  const float* xb  = x  + (size_t)b * C_ * N_;
  const float* xxb = xx + (size_t)b * N_;

  // Preload A fragments for the 32 rows (two 16-row WMMA tiles), K chunks of 4.
  v2f a0[16], a1[16];
  #pragma unroll
  for (int c = 0; c < 16; ++c) {
    int cc = 4 * c + half * 2;
    a0[c].x = xb[(cc + 0) * N_ + r0 + l16];
    a0[c].y = xb[(cc + 1) * N_ + r0 + l16];
    a1[c].x = xb[(cc + 0) * N_ + r0 + 16 + l16];
    a1[c].y = xb[(cc + 1) * N_ + r0 + 16 + l16];
  }
  float xr0[8], xr1[8];
  #pragma unroll
  for (int v = 0; v < 8; ++v) {
    xr0[v] = xxb[r0 + v + 8 * half];
    xr1[v] = xxb[r0 + 16 + v + 8 * half];
  }

  #pragma unroll
  for (int k = 0; k < K_; ++k) { s_val[lane][k] = -__builtin_inff(); s_id[lane][k] = 0; }
  float vmin = -__builtin_inff();
  int   minpos = 0;

  for (int col0 = 0; col0 < N_; col0 += 16) {
    v8f acc0 = {}; v8f acc1 = {};
    #pragma unroll
    for (int c = 0; c < 16; ++c) {
      int cc = 4 * c + half * 2;
      v2f bf;
      bf.x = xb[(cc + 0) * N_ + col0 + l16];
      bf.y = xb[(cc + 1) * N_ + col0 + l16];
      acc0 = __builtin_amdgcn_wmma_f32_16x16x4_f32(false, a0[c], false, bf,
                                                   (short)0, acc0, false, false);
      acc1 = __builtin_amdgcn_wmma_f32_16x16x4_f32(false, a1[c], false, bf,
                                                   (short)0, acc1, false, false);
    }
    float xc = xxb[col0 + l16];
    #pragma unroll
    for (int v = 0; v < 8; ++v) {
      int m = v + 8 * half;
      s_tile[m][l16]      = 2.f * acc0[v] - xr0[v] - xc;
      s_tile[16 + m][l16] = 2.f * acc1[v] - xr1[v] - xc;
    }
    __syncthreads();
    // top-k maintenance: each lane owns one row
    for (int j = 0; j < 16; ++j) {
      float cv = s_tile[lane][j];
      if (cv > vmin) {
        s_val[lane][minpos] = cv;
        s_id[lane][minpos]  = col0 + j;
        vmin = s_val[lane][0]; minpos = 0;
        #pragma unroll
        for (int k = 1; k < K_; ++k) {
          float t = s_val[lane][k];
          if (t < vmin) { vmin = t; minpos = k; }
        }
      }
    }
    __syncthreads();
  }

  const int row = r0 + lane;
  #pragma unroll
  for (int k = 0; k < K_; ++k)
    knn[((size_t)b * N_ + row) * K_ + k] = s_id[lane][k];
}

// ---------------------------------------------------------------------------
// Kernel C: P = W[:, :64] @ x ; Q = (W[:, 64:] - W[:, :64]) @ x  (per batch)
// 64x64x4096 GEMMs via WMMA f32 16x16x4; wave owns one 16-row o-tile, 8 n-tiles.
// ---------------------------------------------------------------------------
__launch_bounds__(32)
__global__ void k_proj(const float* __restrict__ x, const float* __restrict__ W,
                       float* __restrict__ P, float* __restrict__ Q) {
  const int lane = threadIdx.x;
  const int half = lane >> 4;
  const int l16  = lane & 15;
  const int o0   = blockIdx.y * 16;
  const int b    = blockIdx.z;
  const float* xb = x + (size_t)b * C_ * N_;

  v2f aw[16], aq[16];
  #pragma unroll
  for (int c = 0; c < 16; ++c) {
    int cc = 4 * c + half * 2;
    const float* wr = W + (size_t)(o0 + l16) * IN_;
    float w0 = wr[cc + 0], w1 = wr[cc + 1];
    aw[c].x = w0;                    aw[c].y = w1;
    aq[c].x = wr[64 + cc + 0] - w0;  aq[c].y = wr[64 + cc + 1] - w1;
  }

  for (int t = 0; t < 8; ++t) {
    int n0 = (blockIdx.x * 8 + t) * 16;
    v8f p = {}; v8f q = {};
    #pragma unroll
    for (int c = 0; c < 16; ++c) {
      int cc = 4 * c + half * 2;
      v2f bf;
      bf.x = xb[(cc + 0) * N_ + n0 + l16];
      bf.y = xb[(cc + 1) * N_ + n0 + l16];
      p = __builtin_amdgcn_wmma_f32_16x16x4_f32(false, aw[c], false, bf,
                                                (short)0, p, false, false);
      q = __builtin_amdgcn_wmma_f32_16x16x4_f32(false, aq[c], false, bf,
                                                (short)0, q, false, false);
    }
    #pragma unroll
    for (int v = 0; v < 8; ++v) {
      int o = o0 + v + 8 * half;
      size_t off = ((size_t)b * OUT_ + o) * N_ + n0 + l16;
      P[off] = p[v];
      Q[off] = q[v];
    }
  }
}

// ---------------------------------------------------------------------------
// Kernel D: per (b,o): y[n][k] = P[idx[n][k]] + Q[n]; instance-norm stats over
// (n,k), running max over k; max_k commutes with the monotone norm+leakyrelu,
// so output = lrelu((max_k y - mean) * rstd).
// ---------------------------------------------------------------------------
__launch_bounds__(256)
__global__ void k_norm(const float* __restrict__ P, const float* __restrict__ Q,
                       const int* __restrict__ knn, float* __restrict__ out) {
  const int bo  = blockIdx.x;          // b*64 + o
  const int b   = bo >> 6;
  const int tid = threadIdx.x;
  const float* Pr = P + (size_t)bo * N_;
  const float* Qr = Q + (size_t)bo * N_;
  const int*   kb = knn + (size_t)b * N_ * K_;

  float sum = 0.f, sq = 0.f;
  float lm[16];
  #pragma unroll
  for (int i = 0; i < 16; ++i) {
    int n = tid + i * 256;
    float qv = Qr[n];
    const int* id = kb + (size_t)n * K_;
    float mv = -__builtin_inff();
    #pragma unroll
    for (int k = 0; k < K_; ++k) {
      float y = Pr[id[k]] + qv;
      sum += y; sq += y * y;
      mv = fmaxf(mv, y);
    }
    lm[i] = mv;
  }

  __shared__ float rs[256], rq[256];
  rs[tid] = sum; rq[tid] = sq;
  __syncthreads();
  for (int s = 128; s > 0; s >>= 1) {
    if (tid < s) { rs[tid] += rs[tid + s]; rq[tid] += rq[tid + s]; }
    __syncthreads();
  }
  __shared__ float s_mean, s_rstd;
  if (tid == 0) {
    const float inv = 1.f / (float)(N_ * K_);
    float mean = rs[0] * inv;
    float var  = rq[0] * inv - mean * mean;
    s_mean = mean;
    s_rstd = rsqrtf(var + EPS_);
  }
  __syncthreads();
  #pragma unroll
  for (int i = 0; i < 16; ++i) {
    float yn = (lm[i] - s_mean) * s_rstd;
    out[(size_t)bo * N_ + tid + i * 256] = yn >= 0.f ? yn : SLOPE_ * yn;
  }
}

// ---------------------------------------------------------------------------
extern "C" void kernel_launch(void* const* d_in, const int* in_sizes, int n_in,
                              void* d_out, int out_size, void* d_ws, size_t ws_size,
                              hipStream_t stream) {
  const float* x = (const float*)d_in[0];   // (B, C, N) fp32
  const float* W = (const float*)d_in[1];   // (64, 128) fp32
  float* out = (float*)d_out;               // (B, 64, N) fp32

  // workspace layout (~9.3 MB total)
  char* ws = (char*)d_ws;
  float* xx  = (float*)ws;                                   // B*N floats
  int*   knn = (int*)(ws + (size_t)B_ * N_ * 4);             // B*N*K ints
  float* P   = (float*)(ws + (size_t)B_ * N_ * 4 + (size_t)B_ * N_ * K_ * 4);
  float* Q   = P + (size_t)B_ * OUT_ * N_;

  k_xx  <<<(B_ * N_) / 256, 256, 0, stream>>>(x, xx);
  k_knn <<<B_ * (N_ / 32), 32, 0, stream>>>(x, xx, knn);
  k_proj<<<dim3(N_ / (16 * 8), OUT_ / 16, B_), 32, 0, stream>>>(x, W, P, Q);
  k_norm<<<B_ * OUT_, 256, 0, stream>>>(P, Q, knn, out);
}